// HRTAttentionLayer_35338990911501
// MI455X (gfx1250) — compile-verified
//
#include <hip/hip_runtime.h>
#include <hip/hip_bf16.h>

// ---------------------------------------------------------------------------
// HRT attention layer for gfx1250 (MI455X), f32 path via V_WMMA_F32_16X16X4_F32
//   B=2, N=3072, E=4096, NODE_DIM=128, EDGE_DIM=64, H=8, hd=16, QKV=384
//
// GEMM kernels use 2x2 register blocking (32x32 output per wave): each k-step
// issues 8 loads -> 4 WMMAs (2 loads/WMMA, 8 FLOP/B) with distinct
// accumulators (no WMMA->WMMA RAW hazards).
// ---------------------------------------------------------------------------

typedef __attribute__((ext_vector_type(2))) float v2f;
typedef __attribute__((ext_vector_type(8))) float v8f;

#define BSZ        2
#define NN         3072
#define EE         4096
#define NODE_DIM   128
#define EDGE_DIM   64
#define NHEAD      8
#define HDIM       16
#define QKV        384   /* 3*NODE_DIM */

#define WMMA_F32(A, B, C) \
  __builtin_amdgcn_wmma_f32_16x16x4_f32(false, (A), false, (B), (short)0, (C), false, false)

// ---------------------------------------------------------------------------
// 32x32-register-blocked GEMM + bias: C[M,Ncols] = X[M,K] @ W[K,Ncols] + bias
// One wave per 32x32 output tile; K stepped by 4 via f32 WMMA.
// Requires M, Ncols multiples of 32 and K multiple of 4.
// ---------------------------------------------------------------------------
__global__ __launch_bounds__(32)
void gemm_bias_32x32(const float* __restrict__ X, const float* __restrict__ W,
                     const float* __restrict__ bias, float* __restrict__ C,
                     int M, int K, int Ncols) {
  const int tiles_n = Ncols >> 5;
  const int tile = blockIdx.x;
  const int tm = tile / tiles_n;
  const int tn = tile % tiles_n;
  const int lane = threadIdx.x;
  const int half = lane >> 4;     // A/B K sub-index: 0 -> k,k+1   1 -> k+2,k+3
  const int l15  = lane & 15;

  const int row0 = tm * 32 + l15;          // A fragment rows
  const int row1 = row0 + 16;
  const int col0 = tn * 32 + l15;          // B fragment cols
  const int col1 = col0 + 16;

  const float* __restrict__ Xr0 = X + (size_t)row0 * K;
  const float* __restrict__ Xr1 = X + (size_t)row1 * K;

  v8f acc00 = {}, acc01 = {}, acc10 = {}, acc11 = {};
  for (int k = 0; k < K; k += 4) {
    const int ka = k + half * 2;
    v2f a0, a1, b0, b1;
    a0.x = Xr0[ka];                         a0.y = Xr0[ka + 1];
    a1.x = Xr1[ka];                         a1.y = Xr1[ka + 1];
    b0.x = W[(size_t)ka * Ncols + col0];    b0.y = W[(size_t)(ka + 1) * Ncols + col0];
    b1.x = W[(size_t)ka * Ncols + col1];    b1.y = W[(size_t)(ka + 1) * Ncols + col1];
    acc00 = WMMA_F32(a0, b0, acc00);
    acc01 = WMMA_F32(a0, b1, acc01);
    acc10 = WMMA_F32(a1, b0, acc10);
    acc11 = WMMA_F32(a1, b1, acc11);
  }

  const float bv0 = bias[col0];
  const float bv1 = bias[col1];
#pragma unroll
  for (int j = 0; j < 8; ++j) {
    const int r0 = tm * 32 + j + half * 8;  // C/D layout row (upper 16-row band)
    const int r1 = r0 + 16;                 // lower band
    C[(size_t)r0 * Ncols + col0] = acc00[j] + bv0;
    C[(size_t)r0 * Ncols + col1] = acc01[j] + bv1;
    C[(size_t)r1 * Ncols + col0] = acc10[j] + bv0;
    C[(size_t)r1 * Ncols + col1] = acc11[j] + bv1;
  }
}

// ---------------------------------------------------------------------------
// Per-node degree: deg[b,n] = sum_e G[b,e,n]   (coalesced column sums)
// ---------------------------------------------------------------------------
__global__ void degree_kernel(const float* __restrict__ G, float* __restrict__ deg) {
  const int idx = blockIdx.x * blockDim.x + threadIdx.x;   // b*NN + n
  if (idx >= BSZ * NN) return;
  const int b = idx / NN;
  const int n = idx - b * NN;
  const float* __restrict__ g = G + (size_t)b * EE * NN + n;
  float s = 0.f;
  for (int e = 0; e < EE; ++e) s += g[(size_t)e * NN];
  deg[idx] = s;
}

// ---------------------------------------------------------------------------
// Aggregation GEMM + fused combine (2x2 register-blocked, 32x32 per wave):
//   qkv[b,n,c] = qkv_n[b,n,c] + (sum_e G[b,e,n]*qkv_e[b,e,c]) / deg[b,n]
// A = G^T (A[M=n,K=e] = G[e,n], 16-lane-contiguous in n -> coalesced),
// B = qkv_e. K = 4096 stepped by 4. G (100MB) is L2-resident (192MB L2).
// ---------------------------------------------------------------------------
__global__ __launch_bounds__(32)
void agg_kernel(const float* __restrict__ G, const float* __restrict__ qkv_e,
                const float* __restrict__ deg, float* __restrict__ qkv) {
  const int tiles_c = QKV / 32;                 // 12
  const int tiles_n = NN / 32;                  // 96
  int tile = blockIdx.x;
  const int b  = tile / (tiles_n * tiles_c);
  tile -= b * tiles_n * tiles_c;
  const int tn = tile / tiles_c;
  const int tc = tile - tn * tiles_c;

  const int lane = threadIdx.x;
  const int half = lane >> 4;
  const int l15  = lane & 15;
  const int n0 = tn * 32 + l15;
  const int n1 = n0 + 16;
  const int c0 = tc * 32 + l15;
  const int c1 = c0 + 16;

  const float* __restrict__ Gb = G     + (size_t)b * EE * NN;
  const float* __restrict__ Eb = qkv_e + (size_t)b * EE * QKV;

  v8f acc00 = {}, acc01 = {}, acc10 = {}, acc11 = {};
  for (int k = 0; k < EE; k += 4) {
    const int ka = k + half * 2;
    const size_t ga = (size_t)ka * NN;
    const size_t ea = (size_t)ka * QKV;
    v2f a0, a1, b0, b1;
    a0.x = Gb[ga + n0];        a0.y = Gb[ga + NN + n0];
    a1.x = Gb[ga + n1];        a1.y = Gb[ga + NN + n1];
    b0.x = Eb[ea + c0];        b0.y = Eb[ea + QKV + c0];
    b1.x = Eb[ea + c1];        b1.y = Eb[ea + QKV + c1];
    acc00 = WMMA_F32(a0, b0, acc00);
    acc01 = WMMA_F32(a0, b1, acc01);
    acc10 = WMMA_F32(a1, b0, acc10);
    acc11 = WMMA_F32(a1, b1, acc11);
  }

  float* __restrict__ out = qkv + (size_t)b * NN * QKV;
#pragma unroll
  for (int j = 0; j < 8; ++j) {
    const int r0 = tn * 32 + j + half * 8;
    const int r1 = r0 + 16;
    const float rdiv0 = 1.0f / deg[b * NN + r0];
    const float rdiv1 = 1.0f / deg[b * NN + r1];
    const size_t o00 = (size_t)r0 * QKV + c0;
    const size_t o01 = (size_t)r0 * QKV + c1;
    const size_t o10 = (size_t)r1 * QKV + c0;
    const size_t o11 = (size_t)r1 * QKV + c1;
    out[o00] = out[o00] + acc00[j] * rdiv0;
    out[o01] = out[o01] + acc01[j] * rdiv0;
    out[o10] = out[o10] + acc10[j] * rdiv1;
    out[o11] = out[o11] + acc11[j] * rdiv1;
  }
}

// ---------------------------------------------------------------------------
// Flash-style attention, one wave per (b, h, 16-query-row tile), hd = 16.
// qkv layout: [B, N, 384]; head h uses columns [h*48, h*48+48) = q|k|v.
//   S   = (Q*scale) @ K^T       (4x WMMA f32 16x16x4)
//   online softmax over key tiles (shfl_xor butterflies in 16-lane halves)
//   P transposed C-layout -> A-layout through 1KB LDS
//   O  += P @ V                  (4x WMMA)
// ---------------------------------------------------------------------------
__global__ __launch_bounds__(32)
void attn_kernel(const float* __restrict__ qkv, float* __restrict__ attout) {
  __shared__ float lds[16 * 16];

  const int tiles_q = NN / 16;                  // 192
  int tile = blockIdx.x;
  const int b = tile / (NHEAD * tiles_q);
  tile -= b * NHEAD * tiles_q;
  const int h  = tile / tiles_q;
  const int tq = tile - h * tiles_q;

  const int lane = threadIdx.x;
  const int half = lane >> 4;
  const int l15  = lane & 15;
  const float scale = 0.25f;                    // 1/sqrt(16)

  const float* __restrict__ base = qkv + (size_t)b * NN * QKV + h * 48;

  // Q tile in A-layout registers, pre-scaled. qa[2*(kk/4)+{0,1}] = Q[m, kk+half*2+{0,1}]
  float qa[8];
  {
    const int m = tq * 16 + l15;
#pragma unroll
    for (int kk = 0; kk < 16; kk += 4) {
      const int cc = kk + half * 2;
      qa[kk / 2]     = base[(size_t)m * QKV + cc] * scale;
      qa[kk / 2 + 1] = base[(size_t)m * QKV + cc + 1] * scale;
    }
  }

  v8f o = {};
  float mrow[8], lrow[8];
#pragma unroll
  for (int j = 0; j < 8; ++j) { mrow[j] = -1e30f; lrow[j] = 0.f; }

  for (int tk = 0; tk < NN / 16; ++tk) {
    // ---- S = Q @ K^T : B[K=c, N=m] = Kfeat[key m, channel c] ----
    v8f s = {};
    {
      const int m = tk * 16 + l15;              // key index = S column
      const float* __restrict__ krow = base + (size_t)m * QKV + 16;
#pragma unroll
      for (int kk = 0; kk < 16; kk += 4) {
        const int cc = kk + half * 2;
        v2f a, bb;
        a.x  = qa[kk / 2];
        a.y  = qa[kk / 2 + 1];
        bb.x = krow[cc];
        bb.y = krow[cc + 1];
        s = WMMA_F32(a, bb, s);
      }
    }

    // ---- online softmax; row j+8*half lives across the lane's 16-lane half ----
    float pnew[8];
#pragma unroll
    for (int j = 0; j < 8; ++j) {
      const float sj = s[j];
      float mx = sj;
      mx = fmaxf(mx, __shfl_xor(mx, 1, 32));
      mx = fmaxf(mx, __shfl_xor(mx, 2, 32));
      mx = fmaxf(mx, __shfl_xor(mx, 4, 32));
      mx = fmaxf(mx, __shfl_xor(mx, 8, 32));
      const float mnew = fmaxf(mrow[j], mx);
      const float corr = __expf(mrow[j] - mnew);
      const float p    = __expf(sj - mnew);
      float ps = p;
      ps += __shfl_xor(ps, 1, 32);
      ps += __shfl_xor(ps, 2, 32);
      ps += __shfl_xor(ps, 4, 32);
      ps += __shfl_xor(ps, 8, 32);
      lrow[j] = lrow[j] * corr + ps;
      mrow[j] = mnew;
      o[j] *= corr;
      pnew[j] = p;
    }

    // ---- transpose P (C-layout -> A-layout) through LDS ----
#pragma unroll
    for (int j = 0; j < 8; ++j)
      lds[(j + half * 8) * 16 + l15] = pnew[j];
    __syncthreads();   // single-wave workgroup: barrier -> S_NOP, acts as fence

    // ---- O += P @ V ----
#pragma unroll
    for (int kk = 0; kk < 16; kk += 4) {
      const int krow = kk + half * 2;           // key index within tile
      v2f a, bb;
      a.x = lds[l15 * 16 + krow];
      a.y = lds[l15 * 16 + krow + 1];
      const int m = tk * 16 + krow;
      bb.x = base[(size_t)m * QKV + 32 + l15];
      bb.y = base[(size_t)(m + 1) * QKV + 32 + l15];
      o = WMMA_F32(a, bb, o);
    }
    __syncthreads();
  }

  // ---- finalize: divide by row sums, write [B, N, H*16] ----
  float* __restrict__ outp = attout + (size_t)b * NN * NODE_DIM + h * HDIM;
#pragma unroll
  for (int j = 0; j < 8; ++j) {
    const int r = tq * 16 + j + half * 8;
    outp[(size_t)r * NODE_DIM + l15] = o[j] / lrow[j];
  }
}

// ---------------------------------------------------------------------------
// Host-side orchestration
// ---------------------------------------------------------------------------
extern "C" void kernel_launch(void* const* d_in, const int* in_sizes, int n_in,
                              void* d_out, int out_size, void* d_ws, size_t ws_size,
                              hipStream_t stream) {
  const float* node    = (const float*)d_in[0];
  const float* edge    = (const float*)d_in[1];
  const float* G       = (const float*)d_in[2];
  const float* W_qkv_n = (const float*)d_in[3];
  const float* b_qkv_n = (const float*)d_in[4];
  const float* W_qkv_e = (const float*)d_in[5];
  const float* b_qkv_e = (const float*)d_in[6];
  const float* W_o     = (const float*)d_in[7];
  const float* b_o     = (const float*)d_in[8];
  float* out = (float*)d_out;

  // workspace layout (f32): qkv | qkv_e | deg | att   (~25.3 MB total)
  float* qkv  = (float*)d_ws;                        // BSZ*NN*QKV
  float* qkve = qkv  + (size_t)BSZ * NN * QKV;       // BSZ*EE*QKV
  float* deg  = qkve + (size_t)BSZ * EE * QKV;       // BSZ*NN
  float* att  = deg  + (size_t)BSZ * NN;             // BSZ*NN*NODE_DIM

  // 1) qkv_n = node @ W_qkv_n + b   -> qkv (in-place target of combine)
  gemm_bias_32x32<<<(BSZ * NN / 32) * (QKV / 32), 32, 0, stream>>>(
      node, W_qkv_n, b_qkv_n, qkv, BSZ * NN, NODE_DIM, QKV);

  // 2) qkv_e = edge @ W_qkv_e + b
  gemm_bias_32x32<<<(BSZ * EE / 32) * (QKV / 32), 32, 0, stream>>>(
      edge, W_qkv_e, b_qkv_e, qkve, BSZ * EE, EDGE_DIM, QKV);

  // 3) deg[b,n] = sum_e G[b,e,n]
  degree_kernel<<<(BSZ * NN + 255) / 256, 256, 0, stream>>>(G, deg);

  // 4) qkv += (G^T @ qkv_e) / deg
  agg_kernel<<<BSZ * (NN / 32) * (QKV / 32), 32, 0, stream>>>(G, qkve, deg, qkv);

  // 5) per-head flash attention -> att [B, N, 128]
  attn_kernel<<<BSZ * NHEAD * (NN / 16), 32, 0, stream>>>(qkv, att);

  // 6) out = att @ W_o + b_o
  gemm_bias_32x32<<<(BSZ * NN / 32) * (NODE_DIM / 32), 32, 0, stream>>>(
      att, W_o, b_o, out, BSZ * NN, NODE_DIM, NODE_DIM);
}